// LIF_13984413516471
// MI455X (gfx1250) — compile-verified
//
#include <hip/hip_runtime.h>
#include <hip/hip_bf16.h>

// LIF scan: B=128 batches, S=128 seq, H=128 hidden, K=8 thresholds.
// Grid: one block per batch (128 blocks x 128 threads). Thread s owns row s.
// Exact Jacobi fixpoint on per-row carry states, then parallel emit.
//
// Roofline: 8 MiB in + 128 MiB out => ~5.8 us floor @ 23.3 TB/s. Compute is
// ~0.5 GFLOP elementwise (no contraction => WMMA inapplicable); the serial
// T*K=131072 chain is broken by an exact LDS-resident carry fixpoint.
// CDNA5-specific paths used: GLOBAL_LOAD_ASYNC_TO_LDS_B128 + s_wait_asynccnt.

#define LIF_B 128
#define LIF_S 128
#define LIF_H 128
#define LIF_K 8

#define AS_GLOBAL __attribute__((address_space(1)))
#define AS_LDS    __attribute__((address_space(3)))

typedef int lif_v4i __attribute__((ext_vector_type(4)));

#if defined(__HIP_DEVICE_COMPILE__) && __has_builtin(__builtin_amdgcn_global_load_async_to_lds_b128)
#define LIF_HAVE_ASYNC_LDS 1
#pragma message("LIF device pass: using __builtin_amdgcn_global_load_async_to_lds_b128 (ASYNCcnt path)")
#else
#define LIF_HAVE_ASYNC_LDS 0
#if defined(__HIP_DEVICE_COMPILE__)
#pragma message("LIF device pass: async-to-LDS builtin NOT found, falling back to sync LDS staging")
#endif
#endif

#if defined(__HIP_DEVICE_COMPILE__) && __has_builtin(__builtin_amdgcn_s_wait_asynccnt)
#define LIF_WAIT_ASYNC() __builtin_amdgcn_s_wait_asynccnt(0)
#else
#define LIF_WAIT_ASYNC() asm volatile("s_wait_asynccnt 0x0" ::: "memory")
#endif

// Process one (s)-row: 128 time steps x 8 neuron sub-steps. xrow points into LDS.
// Returns carry-out accumulator. EMIT=true additionally writes outs/spikes with
// float4 (b128) stores, contiguous in h for each k.
template <bool EMIT>
__device__ __forceinline__ float lif_row(const float* __restrict__ xrow,
                                         float acc,
                                         const float th[LIF_K],
                                         float* __restrict__ outp,
                                         float* __restrict__ spkp) {
  for (int h0 = 0; h0 < LIF_H; h0 += 4) {
    const float4 xq = *reinterpret_cast<const float4*>(xrow + h0);  // ds_load_b128
    const float xj[4] = {xq.x, xq.y, xq.z, xq.w};
    float ob[LIF_K][4];
    float sb[LIF_K][4];
#pragma unroll
    for (int j = 0; j < 4; ++j) {
      const float x = xj[j];
#pragma unroll
      for (int k = 0; k < LIF_K; ++k) {
        acc += x;                       // s += x_t (re-added every neuron step)
        const bool sp = acc > th[k];    // spike = s > thresh[k]
        const float o = sp ? acc : 0.0f;
        acc -= o;                       // reset to 0 where spiked
        if (EMIT) {
          ob[k][j] = o;
          sb[k][j] = sp ? 1.0f : 0.0f;
        }
      }
    }
    if (EMIT) {
#pragma unroll
      for (int k = 0; k < LIF_K; ++k) {
        *reinterpret_cast<float4*>(outp + k * LIF_H + h0) =
            make_float4(ob[k][0], ob[k][1], ob[k][2], ob[k][3]);
        *reinterpret_cast<float4*>(spkp + k * LIF_H + h0) =
            make_float4(sb[k][0], sb[k][1], sb[k][2], sb[k][3]);
      }
    }
  }
  return acc;
}

__global__ __launch_bounds__(LIF_S) void lif_scan_kernel(
    const float* __restrict__ inp,   // (B,S,H)
    const float* __restrict__ thr,   // (K,)
    float* __restrict__ outs,        // (B,S,K,H)
    float* __restrict__ spks) {      // (B,S,K,H) as 0.0/1.0
  __shared__ float xs[LIF_S * LIF_H];   // 64 KB: this batch's input tile
  __shared__ float carry[LIF_S + 1];    // carry[s] = accumulator entering row s
  __shared__ int changed;

  const int b = blockIdx.x;
  const int tid = threadIdx.x;          // row index s

  // Thresholds are uniform -> scalar loads.
  float th[LIF_K];
#pragma unroll
  for (int k = 0; k < LIF_K; ++k) th[k] = thr[k];

  // ---- Stage inputs[b,:,:] (64 KB) into LDS ----
  const float* gsrc = inp + (size_t)b * (LIF_S * LIF_H);
#if LIF_HAVE_ASYNC_LDS
  {
    AS_GLOBAL lif_v4i* g = (AS_GLOBAL lif_v4i*)gsrc;   // 16B chunks, const dropped
    AS_LDS lif_v4i* l = (AS_LDS lif_v4i*)xs;
    // 4096 x 16B chunks, 128 threads -> 32 async b128 per lane, one wait.
#pragma unroll
    for (int i = 0; i < 32; ++i) {
      const int c = i * LIF_S + tid;
      __builtin_amdgcn_global_load_async_to_lds_b128(g + c, l + c, 0, 0);
    }
    LIF_WAIT_ASYNC();
  }
#else
  for (int i = tid; i < (LIF_S * LIF_H) / 4; i += LIF_S)
    reinterpret_cast<float4*>(xs)[i] = reinterpret_cast<const float4*>(gsrc)[i];
#endif
  if (tid == 0) carry[0] = 0.0f;        // exact initial state
  carry[tid + 1] = 0.0f;                // speculative initial carries
  __syncthreads();

  // ---- Exact Jacobi fixpoint on row carries ----
  // After sweep n, carry[0..n] are exact; an unchanged sweep => global fixpoint
  // of the triangular recurrence => all carries exact. Expected ~2-4 sweeps
  // (reset-to-zero decouples rows as soon as first-spike positions agree);
  // worst case S sweeps == serial cost.
  for (int it = 0; it < LIF_S; ++it) {
    const float cin = carry[tid];
    const float cout = lif_row<false>(&xs[tid * LIF_H], cin, th, nullptr, nullptr);
    __syncthreads();                       // all reads of carry[] complete
    const bool diff = (cout != carry[tid + 1]);
    carry[tid + 1] = cout;
    if (tid == 0) changed = 0;
    __syncthreads();                       // writes + flag-reset visible
    if (diff) changed = 1;                 // benign same-value race
    __syncthreads();                       // flag writes visible
    if (!changed) break;                   // uniform branch
  }

  // ---- Emit pass with converged carries ----
  const float cin = carry[tid];
  const size_t base = ((size_t)b * LIF_S + tid) * (size_t)(LIF_K * LIF_H);
  lif_row<true>(&xs[tid * LIF_H], cin, th, outs + base, spks + base);
}

extern "C" void kernel_launch(void* const* d_in, const int* in_sizes, int n_in,
                              void* d_out, int out_size, void* d_ws, size_t ws_size,
                              hipStream_t stream) {
  (void)in_sizes; (void)n_in; (void)out_size; (void)d_ws; (void)ws_size;
  const float* inp = (const float*)d_in[0];   // (B,S,H) f32
  const float* thr = (const float*)d_in[1];   // (K,)   f32
  float* outs = (float*)d_out;                                        // outs first
  float* spks = outs + (size_t)LIF_B * LIF_S * LIF_K * LIF_H;         // spikes second
  lif_scan_kernel<<<dim3(LIF_B), dim3(LIF_S), 0, stream>>>(inp, thr, outs, spks);
}